// ProjectedCGCraig_89395449299877
// MI455X (gfx1250) — compile-verified
//
#include <hip/hip_runtime.h>

// ---------------------------------------------------------------------------
// Projected-CG + CRAIG solver for MI455X (gfx1250), wave32.
//  * all matvecs on V_WMMA_F32_16X16X4_F32 (full fp32 matrix pipe)
//  * one persistent kernel per CRAIG solve (grid barriers, deterministic
//    fixed-order cross-block reductions) -> ~135 launches total vs ~4500
//  * v2f (b64) loads in WMMA loops, global_prefetch for streaming B
// ---------------------------------------------------------------------------

typedef __attribute__((ext_vector_type(2))) float v2f;
typedef __attribute__((ext_vector_type(8))) float v8f;

#define CRAIG_TOL_F 1e-10f
#define CG_TOL2_F   1e-16f   // CG_TOL^2
#define CG_REG_F    1e-06f
#define BRK_TOL_F   1e-14f
#define TINY_F      1e-30f

#define PK_BLOCKS  32
#define PK_THREADS 256

// ---------------------------------------------------------------------------
// block / grid synchronization primitives
// ---------------------------------------------------------------------------
__device__ __forceinline__ float block_reduce_sum(float val, float* red) {
  int t = threadIdx.x;
  red[t] = val;
  __syncthreads();
  for (int s = blockDim.x >> 1; s > 0; s >>= 1) {
    if (t < s) red[t] += red[t + s];
    __syncthreads();
  }
  float r = red[0];
  __syncthreads();
  return r;
}

// sense-reversing grid barrier; bar[0]=arrival count, bar[1]=generation
__device__ __forceinline__ void gsync(unsigned* bar) {
  __threadfence();                               // release prior writes
  __syncthreads();
  if (threadIdx.x == 0) {
    unsigned gen = __hip_atomic_load(&bar[1], __ATOMIC_RELAXED,
                                     __HIP_MEMORY_SCOPE_AGENT);
    unsigned arrived = __hip_atomic_fetch_add(&bar[0], 1u, __ATOMIC_ACQ_REL,
                                              __HIP_MEMORY_SCOPE_AGENT) + 1u;
    if (arrived == (unsigned)PK_BLOCKS) {
      __hip_atomic_store(&bar[0], 0u, __ATOMIC_RELAXED, __HIP_MEMORY_SCOPE_AGENT);
      __hip_atomic_store(&bar[1], gen + 1u, __ATOMIC_RELEASE,
                         __HIP_MEMORY_SCOPE_AGENT);
    } else {
      while (__hip_atomic_load(&bar[1], __ATOMIC_ACQUIRE,
                               __HIP_MEMORY_SCOPE_AGENT) == gen)
        __builtin_amdgcn_s_sleep(1);
    }
  }
  __syncthreads();
  __threadfence();                               // acquire: see peers' writes
}

// deterministic grid-wide sum: per-block partials into a fresh slot group,
// every thread re-sums the 32 partials in fixed order (bitwise reproducible).
__device__ __forceinline__ float grid_reduce(float part, float* red,
                                             float* slots, int& sidx,
                                             unsigned* bar) {
  float bsum = block_reduce_sum(part, red);
  float* slot = slots + (size_t)sidx * PK_BLOCKS;
  if (threadIdx.x == 0) slot[blockIdx.x] = bsum;
  ++sidx;
  gsync(bar);
  float r = 0.0f;
#pragma unroll
  for (int j = 0; j < PK_BLOCKS; ++j) r += slot[j];
  return r;
}

// ---------------------------------------------------------------------------
// WMMA matvec bodies (device functions used inside the persistent kernel)
// ---------------------------------------------------------------------------
// t[n] = sum_m A[m,n]*u[m]  (reduction over rows; coalesced 4x16 B-frags)
__device__ __forceinline__ void mv_colred(const float* __restrict__ A,
                                          const float* __restrict__ u,
                                          float* __restrict__ out,
                                          int Mrows, int Ncols, float* sU) {
  for (int i = threadIdx.x; i < Mrows; i += blockDim.x) sU[i] = u[i];
  __syncthreads();
  const int lane  = threadIdx.x & 31;
  const int wave  = threadIdx.x >> 5;
  const int gwave = blockIdx.x * (blockDim.x >> 5) + wave;
  const int n0    = gwave * 16;
  if (n0 < Ncols) {
    const int k0   = (lane >> 4) * 2;            // 0 | 2
    const int ncol = n0 + (lane & 15);
    v8f c = {};
    for (int m0 = 0; m0 < Mrows; m0 += 4) {
      v2f a = *(const v2f*)(sU + m0 + k0);       // u broadcast (ds_load_b64)
      v2f b;
      b.x = A[(size_t)(m0 + k0) * Ncols + ncol]; // 4x16 tile rows (coalesced)
      b.y = A[(size_t)(m0 + k0 + 1) * Ncols + ncol];
      c = __builtin_amdgcn_wmma_f32_16x16x4_f32(false, a, false, b,
                                                (short)0, c, false, false);
    }
    if (lane < 16) out[n0 + lane] = c[0];        // row 0 of D
  }
  __syncthreads();                               // smem reused by caller
}

// y[m] = sum_n M[m,n]*x[n]  (reduction over cols; 16x4 A-frags, x in LDS)
__device__ __forceinline__ void mv_row(const float* __restrict__ Mmat,
                                       const float* __restrict__ x,
                                       float* __restrict__ out,
                                       int Rrows, int Ccols, float* sX) {
  const int wavesPerBlock = blockDim.x >> 5;
  const int rowWaves = Rrows >> 4;
  if (blockIdx.x * wavesPerBlock >= rowWaves) return;   // idle block
  for (int i = threadIdx.x; i < Ccols; i += blockDim.x) sX[i] = x[i];
  __syncthreads();
  const int lane  = threadIdx.x & 31;
  const int gwave = blockIdx.x * wavesPerBlock + (threadIdx.x >> 5);
  if (gwave < rowWaves) {
    const int m0 = gwave * 16;
    const int k0 = (lane >> 4) * 2;
    const float* rowp = Mmat + (size_t)(m0 + (lane & 15)) * Ccols;
    v8f c = {};
    for (int n0 = 0; n0 < Ccols; n0 += 4) {
      v2f a = *(const v2f*)(rowp + n0 + k0);     // global_load_b64
      v2f b = *(const v2f*)(sX + n0 + k0);       // ds_load_b64 (broadcast)
      c = __builtin_amdgcn_wmma_f32_16x16x4_f32(false, a, false, b,
                                                (short)0, c, false, false);
    }
    if (lane == 0) {
#pragma unroll
      for (int r = 0; r < 8; ++r) out[m0 + r] = c[r];
    } else if (lane == 16) {
#pragma unroll
      for (int r = 0; r < 8; ++r) out[m0 + 8 + r] = c[r];
    }
  }
  __syncthreads();
}

// ---------------------------------------------------------------------------
// persistent kernel: one full CRAIG solve (init + 40 flag-frozen iterations)
// all convergence scalars are recomputed identically by every thread from
// grid-reduced values -> uniform control flow, no flag traffic.
// ---------------------------------------------------------------------------
__global__ __launch_bounds__(PK_THREADS)
void k_craig_persistent(const float* __restrict__ A, int M_, int N_,
                        const float* __restrict__ rhs,
                        const float* __restrict__ parent,
                        float* __restrict__ u, float* __restrict__ v,
                        float* __restrict__ xc, float* __restrict__ t4,
                        float* __restrict__ w5,
                        float* __restrict__ slots, unsigned* __restrict__ bar) {
  __shared__ float red[PK_THREADS];
  __shared__ float smem[4096];                   // staging (N <= 4096)
  if (parent[0] != 0.0f) return;                 // CG frozen: whole solve skips

  const int gtid = blockIdx.x * blockDim.x + threadIdx.x;
  const int NT   = gridDim.x * blockDim.x;
  int sidx = 0;

  // ---- init: beta1, u = rhs / beta1_safe ---------------------------------
  float part = 0.0f;
  for (int i = gtid; i < M_; i += NT) { float rv = rhs[i]; part += rv * rv; }
  const float beta1 = sqrtf(grid_reduce(part, red, slots, sidx, bar));
  const float b1s   = fmaxf(beta1, TINY_F);
  for (int i = gtid; i < M_; i += NT) u[i] = rhs[i] / b1s;
  gsync(bar);

  mv_colred(A, u, t4, M_, N_, smem);             // t4 = A^T u
  gsync(bar);

  part = 0.0f;
  for (int i = gtid; i < N_; i += NT) { float t = t4[i]; part += t * t; }
  float alpha = sqrtf(grid_reduce(part, red, slots, sidx, bar));
  const int brk_init = alpha < BRK_TOL_F;
  float asafe = fmaxf(alpha, TINY_F);
  float s = beta1 / asafe;
  for (int i = gtid; i < N_; i += NT) {
    float vv = t4[i] / asafe;
    v[i] = vv;
    xc[i] = brk_init ? 0.0f : s * vv;
  }
  gsync(bar);

  mv_row(A, v, w5, M_, N_, smem);                // w5 = A v
  gsync(bar);

  part = 0.0f;
  for (int i = gtid; i < M_; i += NT) {
    float uh = w5[i] - alpha * u[i];
    part += uh * uh;
  }
  float beta = sqrtf(grid_reduce(part, red, slots, sidx, bar));
  float bsafe = fmaxf(beta, TINY_F);
  for (int i = gtid; i < M_; i += NT) u[i] = (w5[i] - alpha * u[i]) / bsafe;
  const int triv = beta1 < CRAIG_TOL_F * fmaxf(b1s, 1.0f);
  float resid = fabsf(beta * s);
  int conv = triv || (resid < CRAIG_TOL_F * b1s);
  int done = conv || (brk_init && !triv);
  gsync(bar);

  // ---- 40 Golub-Kahan iterations (uniform break == lax.cond freeze) ------
  for (int it = 0; it < 40 && !done; ++it) {
    mv_colred(A, u, t4, M_, N_, smem);           // t4 = A^T u
    gsync(bar);

    // v_hat, reorthogonalize against v, normalize, update x
    part = 0.0f;
    for (int i = gtid; i < N_; i += NT) {
      float vh = t4[i] - beta * v[i];
      t4[i] = vh;
      part += v[i] * vh;
    }
    const float dvv = grid_reduce(part, red, slots, sidx, bar);
    part = 0.0f;
    for (int i = gtid; i < N_; i += NT) {
      float vh2 = t4[i] - dvv * v[i];
      t4[i] = vh2;
      part += vh2 * vh2;
    }
    const float alpha_new = sqrtf(grid_reduce(part, red, slots, sidx, bar));
    const int brkA = alpha_new < BRK_TOL_F;
    asafe = fmaxf(alpha_new, TINY_F);
    const float s_new = -beta * s / asafe;
    for (int i = gtid; i < N_; i += NT) {
      float vn = t4[i] / asafe;
      v[i] = vn;
      if (!brkA) xc[i] += s_new * vn;
    }
    alpha = alpha_new; s = s_new;
    gsync(bar);

    mv_row(A, v, w5, M_, N_, smem);              // w5 = A v
    gsync(bar);

    // u_hat, reorthogonalize against u, normalize
    part = 0.0f;
    for (int i = gtid; i < M_; i += NT)
      part += u[i] * (w5[i] - alpha * u[i]);
    const float duu = grid_reduce(part, red, slots, sidx, bar);
    part = 0.0f;
    for (int i = gtid; i < M_; i += NT) {
      float uh2 = (w5[i] - alpha * u[i]) - duu * u[i];
      part += uh2 * uh2;
    }
    const float beta_new = sqrtf(grid_reduce(part, red, slots, sidx, bar));
    const int brkB = beta_new < BRK_TOL_F;
    bsafe = fmaxf(beta_new, TINY_F);
    for (int i = gtid; i < M_; i += NT)
      u[i] = ((w5[i] - alpha * u[i]) - duu * u[i]) / bsafe;
    beta = beta_new;
    resid = fabsf(beta_new * s);
    conv = resid < CRAIG_TOL_F * b1s;
    done = conv || brkA || brkB;
    gsync(bar);
  }
}

// ---------------------------------------------------------------------------
// standalone WMMA matvec (B @ p streaming, projection pre-matvecs); guarded
// ---------------------------------------------------------------------------
__global__ void k_matvec_row(const float* __restrict__ Mmat,
                             const float* __restrict__ xvec,
                             float* __restrict__ out,
                             int Rrows, int Ccols,
                             const float* __restrict__ guard) {
  if (guard[0] != 0.0f) return;
  extern __shared__ float sX[];
  for (int i = threadIdx.x; i < Ccols; i += blockDim.x) sX[i] = xvec[i];
  __syncthreads();
  const int lane  = threadIdx.x & 31;
  const int gwave = blockIdx.x * (blockDim.x >> 5) + (threadIdx.x >> 5);
  const int m0    = gwave * 16;
  if (m0 >= Rrows) return;
  const int k0 = (lane >> 4) * 2;
  const float* rowp = Mmat + (size_t)(m0 + (lane & 15)) * Ccols;
  v8f c = {};
  for (int nb = 0; nb < Ccols; nb += 512) {
    __builtin_prefetch(rowp + nb + 512, 0, 1);   // global_prefetch: prime L2
    const int nend = (nb + 512 < Ccols) ? nb + 512 : Ccols;
    for (int n0 = nb; n0 < nend; n0 += 4) {
      v2f a = *(const v2f*)(rowp + n0 + k0);
      v2f b = *(const v2f*)(sX + n0 + k0);
      c = __builtin_amdgcn_wmma_f32_16x16x4_f32(false, a, false, b,
                                                (short)0, c, false, false);
    }
  }
  if (lane == 0) {
#pragma unroll
    for (int r = 0; r < 8; ++r) out[m0 + r] = c[r];
  } else if (lane == 16) {
#pragma unroll
    for (int r = 0; r < 8; ++r) out[m0 + 8 + r] = c[r];
  }
}

// ---------------------------------------------------------------------------
// CG kernels (single block, deterministic LDS reductions)
// ---------------------------------------------------------------------------
__global__ void k_zero_scalars(float* S) { S[threadIdx.x] = 0.0f; }

__global__ void k_cg_init(float* __restrict__ S, const float* __restrict__ r,
                          float* __restrict__ d, float* __restrict__ p, int N_) {
  __shared__ float red[1024];
  float acc = 0.0f;
  for (int i = threadIdx.x; i < N_; i += blockDim.x) {
    float rv = r[i];
    acc += rv * rv;
    d[i] = 0.0f;
    p[i] = rv;
  }
  float rz = block_reduce_sum(acc, red);
  if (threadIdx.x == 0) { S[16] = rz; S[19] = (rz < CG_TOL2_F) ? 1.0f : 0.0f; }
}

__global__ void k_cg_update(float* __restrict__ S, float* __restrict__ d,
                            float* __restrict__ r, float* __restrict__ p,
                            const float* __restrict__ pbp, int N_) {
  __shared__ float red[1024];
  if (S[19] != 0.0f) return;
  float acc1 = 0.0f, acc2 = 0.0f;
  for (int i = threadIdx.x; i < N_; i += blockDim.x) {
    float pi = p[i];
    acc1 += pi * pbp[i];
    acc2 += pi * pi;
  }
  float pPBp = block_reduce_sum(acc1, red);
  float pp   = block_reduce_sum(acc2, red);
  if (pPBp <= CG_REG_F * pp) {                   // curvature guard: freeze
    if (threadIdx.x == 0) S[19] = 1.0f;
    return;
  }
  float rz = S[16];
  float alpha = rz / fmaxf(pPBp, TINY_F);
  float acc3 = 0.0f;
  for (int i = threadIdx.x; i < N_; i += blockDim.x) {
    d[i] += alpha * p[i];
    float rn = r[i] - alpha * pbp[i];
    r[i] = rn;
    acc3 += rn * rn;
  }
  float rn2 = block_reduce_sum(acc3, red);
  float beta = rn2 / fmaxf(rz, TINY_F);
  for (int i = threadIdx.x; i < N_; i += blockDim.x)
    p[i] = r[i] + beta * p[i];
  if (threadIdx.x == 0) { S[16] = rn2; S[19] = (rn2 < CG_TOL2_F) ? 1.0f : 0.0f; }
}

// ---------------------------------------------------------------------------
// small vector kernels
// ---------------------------------------------------------------------------
__global__ void k_copy(float* __restrict__ dst, const float* __restrict__ src, int n) {
  int i = blockIdx.x * blockDim.x + threadIdx.x;
  if (i < n) dst[i] = src[i];
}
__global__ void k_neg(float* __restrict__ dst, const float* __restrict__ src, int n) {
  int i = blockIdx.x * blockDim.x + threadIdx.x;
  if (i < n) dst[i] = -src[i];
}
__global__ void k_epilogue(float* __restrict__ out, const float* __restrict__ in,
                           const float* __restrict__ xc,
                           const float* __restrict__ guard, int n) {
  if (guard[0] != 0.0f) return;
  int i = blockIdx.x * blockDim.x + threadIdx.x;
  if (i < n) out[i] = in[i] - xc[i];
}

// ---------------------------------------------------------------------------
extern "C" void kernel_launch(void* const* d_in, const int* in_sizes, int n_in,
                              void* d_out, int out_size, void* d_ws, size_t ws_size,
                              hipStream_t stream) {
  (void)n_in; (void)out_size; (void)ws_size;
  const float* A = (const float*)d_in[0];   // (M, N) row-major
  const float* b = (const float*)d_in[1];   // (M,)
  const float* B = (const float*)d_in[2];   // (N, N)
  const float* g = (const float*)d_in[3];   // (N,)
  float* out = (float*)d_out;               // [x (N) | d (N)]
  const int M_ = in_sizes[1];               // 512
  const int N_ = in_sizes[3];               // 4096

  float*    W     = (float*)d_ws;
  float*    S     = W;                 // 64 scalars (S16=rz, S19=cg flag, S63=0)
  unsigned* bar   = (unsigned*)(W + 32);  // barrier state (S[32..33], zeroed)
  float*    u     = W + 64;            // M
  float*    v     = W + 1024;          // N
  float*    xc    = W + 5120;          // N  (CRAIG x)
  float*    t4    = W + 9216;          // N
  float*    w5    = W + 13312;         // M  (A v / rhs staging)
  float*    dv    = W + 14336;         // N  (CG d)
  float*    rv    = W + 18432;         // N  (CG r)
  float*    pv    = W + 22528;         // N  (CG p)
  float*    Bp    = W + 26624;         // N  (B @ p)
  float*    pbp   = W + 30720;         // N  (projected B p)
  float*    pin   = W + 34816;         // N  (-g staging)
  float*    slots = W + 38912;         // 6144 (reduction slot pool)

  k_zero_scalars<<<1, 64, 0, stream>>>(S);
  const float* ZERO = S + 63;
  const int vb = (N_ + 255) / 256;
  const size_t shRow = (size_t)N_ * sizeof(float);

  // ---- x = craig_solve(A, b).x -------------------------------------------
  k_craig_persistent<<<PK_BLOCKS, PK_THREADS, 0, stream>>>(
      A, M_, N_, b, ZERO, u, v, xc, t4, w5, slots, bar);
  k_copy<<<vb, 256, 0, stream>>>(out, xc, N_);

  // ---- r0 = project(-g) = -g - craig(A, A@(-g)).x ------------------------
  k_neg<<<vb, 256, 0, stream>>>(pin, g, N_);
  k_matvec_row<<<M_ / 128, 256, shRow, stream>>>(A, pin, w5, M_, N_, ZERO);
  k_craig_persistent<<<PK_BLOCKS, PK_THREADS, 0, stream>>>(
      A, M_, N_, w5, ZERO, u, v, xc, t4, w5, slots, bar);
  k_epilogue<<<vb, 256, 0, stream>>>(rv, pin, xc, ZERO, N_);
  k_cg_init<<<1, 1024, 0, stream>>>(S, rv, dv, pv, N_);

  // ---- 25 projected-CG steps (state frozen via S[19] once converged) -----
  for (int it = 0; it < 25; ++it) {
    k_matvec_row<<<N_ / 128, 256, shRow, stream>>>(B, pv, Bp, N_, N_, S + 19);
    k_matvec_row<<<M_ / 128, 256, shRow, stream>>>(A, Bp, w5, M_, N_, S + 19);
    k_craig_persistent<<<PK_BLOCKS, PK_THREADS, 0, stream>>>(
        A, M_, N_, w5, S + 19, u, v, xc, t4, w5, slots, bar);
    k_epilogue<<<vb, 256, 0, stream>>>(pbp, Bp, xc, S + 19, N_);
    k_cg_update<<<1, 1024, 0, stream>>>(S, dv, rv, pv, pbp, N_);
  }
  k_copy<<<vb, 256, 0, stream>>>(out + N_, dv, N_);
}